// FractalNet_40570261078561
// MI455X (gfx1250) — compile-verified
//
#include <hip/hip_runtime.h>

// CDNA5 / gfx1250, wave32. Full-fp32 WMMA path (V_WMMA_F32_16X16X4_F32)
// + async global->LDS staging of the A strip (ASYNCcnt path).

typedef float v2f __attribute__((ext_vector_type(2)));
typedef float v8f __attribute__((ext_vector_type(8)));

static constexpr int NN = 256;   // N nodes / matrix dim
static constexpr int HD = 64;    // hidden
static constexpr int CC = 32;    // colors
static constexpr int SP = 260;   // padded LDS row stride (dwords): banks 4m+c

__device__ __forceinline__ v8f wmma_f32x4(v2f a, v2f b, v8f c) {
  // 8 args: (neg_a, A, neg_b, B, c_mod, C, reuse_a, reuse_b)
  return __builtin_amdgcn_wmma_f32_16x16x4_f32(false, a, false, b, (short)0, c,
                                               false, false);
}

// Async copy of one 16-row A strip (16KB) into padded LDS tile.
// nthreads must divide 1024; each 16B chunk id: row = id>>6, col16 = id&63.
template <int NTHREADS>
__device__ __forceinline__ void async_stage_strip(const float* __restrict__ gbase,
                                                  float* __restrict__ tile) {
  const int t = threadIdx.x;
#pragma unroll
  for (int c = 0; c < 1024 / NTHREADS; ++c) {
    int id    = c * NTHREADS + t;      // 0..1023 (16B chunks)
    int row   = id >> 6;               // 0..15
    int col16 = id & 63;               // 16B units within row
    unsigned long long ga =
        (unsigned long long)(uintptr_t)gbase + (unsigned)(row * (NN * 4) + col16 * 16);
    unsigned la = (unsigned)(uintptr_t)(tile + row * SP) + (unsigned)(col16 * 16);
    asm volatile("global_load_async_to_lds_b128 %0, %1, off"
                 :
                 : "v"(la), "v"(ga)
                 : "memory");
  }
  asm volatile("s_wait_asynccnt 0x0" ::: "memory");
  __syncthreads();
}

// ---------------------------------------------------------------------------
// Kernel 1: per-node MLP + softmax -> L (blockIdx.x < 256) or R (>= 256)
// ---------------------------------------------------------------------------
__global__ __launch_bounds__(256) void probs_kernel(
    const float* __restrict__ W1L, const float* __restrict__ b1L,
    const float* __restrict__ W2L, const float* __restrict__ b2L,
    const float* __restrict__ W3L, const float* __restrict__ b3L,
    const float* __restrict__ W1R, const float* __restrict__ b1R,
    const float* __restrict__ W2R, const float* __restrict__ b2R,
    const float* __restrict__ W3R, const float* __restrict__ b3R,
    float* __restrict__ Lout, float* __restrict__ Rout) {
  __shared__ float h1[HD];
  __shared__ float h2[HD];
  __shared__ float red[NN];

  const int n     = blockIdx.x & (NN - 1);
  const int right = blockIdx.x >> 8;
  const float* W1 = right ? W1R : W1L;
  const float* b1 = right ? b1R : b1L;
  const float* W2 = right ? W2R : W2L;
  const float* b2 = right ? b2R : b2L;
  const float* W3 = right ? W3R : W3L;
  const float* b3 = right ? b3R : b3L;
  float* P        = right ? Rout : Lout;

  const int t = threadIdx.x;

  if (t < HD) {
    float v = W1[n * HD + t] + b1[n * HD + t];
    h1[t] = v > 0.f ? v : 0.f;
  }
  __syncthreads();

  if (t < HD) {
    const float* w2 = W2 + (size_t)(n * HD + t) * HD;
    float acc = b2[n * HD + t];
#pragma unroll 8
    for (int h = 0; h < HD; ++h) acc += h1[h] * w2[h];
    h2[t] = acc > 0.f ? acc : 0.f;
  }
  __syncthreads();

  const float* w3 = W3 + (size_t)(n * NN + t) * HD;
  float logit = b3[n * NN + t];
#pragma unroll 8
  for (int k = 0; k < HD; ++k) logit += h2[k] * w3[k];

  red[t] = logit;
  __syncthreads();
  for (int s = 128; s > 0; s >>= 1) {
    if (t < s) red[t] = fmaxf(red[t], red[t + s]);
    __syncthreads();
  }
  const float mx = red[0];
  __syncthreads();
  const float e = __expf(logit - mx);
  red[t] = e;
  __syncthreads();
  for (int s = 128; s > 0; s >>= 1) {
    if (t < s) red[t] += red[t + s];
    __syncthreads();
  }
  P[(size_t)n * NN + t] = e / red[0];
}

// ---------------------------------------------------------------------------
// Kernel 2: seed A after depth 0: row 0 = L[0,:], row 1 = R[0,:]
// ---------------------------------------------------------------------------
__global__ __launch_bounds__(256) void init_kernel(const float* __restrict__ Lm,
                                                   const float* __restrict__ Rm,
                                                   float* __restrict__ A) {
  int t = blockIdx.x * 256 + threadIdx.x;  // 0..511
  if (t < NN)
    A[t] = Lm[t];
  else
    A[t] = Rm[t - NN];
}

// ---------------------------------------------------------------------------
// Kernel 3: LC = L @ colors, RC = R @ colors   (each 256x32)
// ---------------------------------------------------------------------------
__global__ __launch_bounds__(256) void lc_kernel(const float* __restrict__ Lm,
                                                 const float* __restrict__ Rm,
                                                 const float* __restrict__ colors,
                                                 float* __restrict__ LC,
                                                 float* __restrict__ RC) {
  int tid = blockIdx.x * 256 + threadIdx.x;  // 0..16383
  int mat = tid >> 13;
  int rem = tid & 8191;
  int a   = rem >> 5;
  int c   = rem & 31;
  const float* P = mat ? Rm : Lm;
  float acc = 0.f;
#pragma unroll 8
  for (int b = 0; b < NN; ++b) acc += P[(size_t)a * NN + b] * colors[b * CC + c];
  (mat ? RC : LC)[a * CC + c] = acc;
}

// ---------------------------------------------------------------------------
// Kernel 4a: guarded expansion for tiny M (2/4/8).  Runtime mode, row guards.
// ---------------------------------------------------------------------------
__global__ __launch_bounds__(256) void expand_small(
    const float* __restrict__ Ain, float* __restrict__ Aout,
    const float* __restrict__ Lm, const float* __restrict__ Rm,
    int M, int wshift, int mode) {
  const int lane = threadIdx.x & 31;
  const int wave = threadIdx.x >> 5;
  const int task = blockIdx.y * 8 + wave;  // 0..31
  const int mat  = task >> 4;
  const int n0   = (task & 15) << 4;

  const float* __restrict__ B = mat ? Rm : Lm;

  const int m    = lane & 15;
  const int koff = (lane >> 4) << 1;
  int arow = m;
  if (arow > M - 1) arow = M - 1;
  const float* aptr = Ain + (size_t)arow * NN + koff;
  const float* bptr = B + (size_t)koff * NN + n0 + m;

  v8f acc = {0.f, 0.f, 0.f, 0.f, 0.f, 0.f, 0.f, 0.f};
#pragma unroll 4
  for (int k0 = 0; k0 < NN; k0 += 4) {
    v2f a = *(const v2f*)(aptr + k0);
    v2f b;
    b.x = bptr[k0 * NN];
    b.y = bptr[k0 * NN + NN];
    acc = wmma_f32x4(a, b, acc);
  }

  const int base_m = (lane >> 4) << 3;
  const int nidx   = n0 + m;
  const int Wm1    = (1 << wshift) - 1;
#pragma unroll
  for (int i = 0; i < 8; ++i) {
    int r = base_m + i;
    if (r < M) {
      int dst;
      if (mode) {
        int h = r >> wshift;
        int w = r & Wm1;
        dst = (((h << 1) + mat) << wshift) + w;
      } else {
        dst = (r << 1) + mat;
      }
      Aout[(size_t)dst * NN + nidx] = acc[i];
    }
  }
}

// ---------------------------------------------------------------------------
// Kernel 4b: fast expansion (M multiple of 16, W >= 8).  MODE: 0 even, 1 odd.
// Block = 8 waves; A strip staged in LDS via async copy; one 16x16 tile/wave.
// ---------------------------------------------------------------------------
template <int MODE>
__global__ __launch_bounds__(256) void expand_fast(
    const float* __restrict__ Ain, float* __restrict__ Aout,
    const float* __restrict__ Lm, const float* __restrict__ Rm,
    int wshift) {
  __shared__ float tileA[16 * SP];

  const int lane = threadIdx.x & 31;
  const int wave = threadIdx.x >> 5;
  const int task = blockIdx.y * 8 + wave;  // 0..31
  const int mat  = task >> 4;              // 0 = L, 1 = R
  const int n0   = (task & 15) << 4;
  const int row0 = blockIdx.x << 4;

  async_stage_strip<256>(Ain + (size_t)row0 * NN, tileA);

  const float* __restrict__ B = mat ? Rm : Lm;

  const int m    = lane & 15;
  const int koff = (lane >> 4) << 1;
  const float* aptr = tileA + m * SP + koff;       // LDS, padded rows
  const float* bptr = B + (size_t)koff * NN + n0 + m;

  v8f acc = {0.f, 0.f, 0.f, 0.f, 0.f, 0.f, 0.f, 0.f};
#pragma unroll 4
  for (int k0 = 0; k0 < NN; k0 += 4) {
    v2f a = *(const v2f*)(aptr + k0);               // ds_load_b64
    v2f b;
    b.x = bptr[k0 * NN];
    b.y = bptr[k0 * NN + NN];
    acc = wmma_f32x4(a, b, acc);
  }

  // Hoisted destination: lane's 8 rows share h (W >= 8, 8-aligned block).
  const int r0 = row0 + ((lane >> 4) << 3);
  int dst0;
  if (MODE) {
    int h = r0 >> wshift;
    int w = r0 & ((1 << wshift) - 1);
    dst0 = (((h << 1) + mat) << wshift) + w;        // stride 1 per row
  } else {
    dst0 = (r0 << 1) + mat;                          // stride 2 per row
  }
  float* op = Aout + (size_t)dst0 * NN + (n0 + m);
  constexpr int STRIDE = MODE ? 1 : 2;
#pragma unroll
  for (int i = 0; i < 8; ++i) op[(size_t)(i * STRIDE) * NN] = acc[i];
}

// ---------------------------------------------------------------------------
// Kernel 5: fused last depth + colors: out = A @ {LC|RC}, odd-mode scatter.
// Block = 4 waves; A strip staged in LDS via async copy.
// ---------------------------------------------------------------------------
__global__ __launch_bounds__(128) void final_kernel(
    const float* __restrict__ Ain, const float* __restrict__ LC,
    const float* __restrict__ RC, float* __restrict__ out, int wshift) {
  __shared__ float tileA[16 * SP];

  const int lane = threadIdx.x & 31;
  const int wave = threadIdx.x >> 5;  // 0..3
  const int mat  = wave >> 1;
  const int n0   = (wave & 1) << 4;
  const int row0 = blockIdx.x << 4;

  async_stage_strip<128>(Ain + (size_t)row0 * NN, tileA);

  const float* __restrict__ B = mat ? RC : LC;

  const int m    = lane & 15;
  const int koff = (lane >> 4) << 1;
  const float* aptr = tileA + m * SP + koff;
  const float* bptr = B + koff * CC + n0 + m;

  v8f acc = {0.f, 0.f, 0.f, 0.f, 0.f, 0.f, 0.f, 0.f};
#pragma unroll 4
  for (int k0 = 0; k0 < NN; k0 += 4) {
    v2f a = *(const v2f*)(aptr + k0);
    v2f b;
    b.x = bptr[k0 * CC];
    b.y = bptr[k0 * CC + CC];
    acc = wmma_f32x4(a, b, acc);
  }

  const int r0 = row0 + ((lane >> 4) << 3);
  int h = r0 >> wshift;
  int w = r0 & ((1 << wshift) - 1);
  int dst0 = (((h << 1) + mat) << wshift) + w;      // stride 1 per row
  float* op = out + (size_t)dst0 * CC + (n0 + m);
#pragma unroll
  for (int i = 0; i < 8; ++i) op[i * CC] = acc[i];
}

// ---------------------------------------------------------------------------
// Workspace (floats): L,R (65536 ea), LC,RC (8192 ea), A0,A1 (32768*256 ea)
// total ~64.6 MB.
// ---------------------------------------------------------------------------
extern "C" void kernel_launch(void* const* d_in, const int* in_sizes, int n_in,
                              void* d_out, int out_size, void* d_ws,
                              size_t ws_size, hipStream_t stream) {
  (void)in_sizes; (void)n_in; (void)out_size; (void)ws_size;
  const float* W1L = (const float*)d_in[0];
  const float* b1L = (const float*)d_in[1];
  const float* W2L = (const float*)d_in[2];
  const float* b2L = (const float*)d_in[3];
  const float* W3L = (const float*)d_in[4];
  const float* b3L = (const float*)d_in[5];
  const float* W1R = (const float*)d_in[6];
  const float* b1R = (const float*)d_in[7];
  const float* W2R = (const float*)d_in[8];
  const float* b2R = (const float*)d_in[9];
  const float* W3R = (const float*)d_in[10];
  const float* b3R = (const float*)d_in[11];
  const float* colors = (const float*)d_in[12];
  float* out = (float*)d_out;

  float* ws = (float*)d_ws;
  float* Lm = ws;
  float* Rm = Lm + (size_t)NN * NN;
  float* LC = Rm + (size_t)NN * NN;
  float* RC = LC + (size_t)NN * CC;
  float* A0 = RC + (size_t)NN * CC;
  float* A1 = A0 + (size_t)32768 * NN;

  probs_kernel<<<dim3(512), dim3(256), 0, stream>>>(
      W1L, b1L, W2L, b2L, W3L, b3L, W1R, b1R, W2R, b2R, W3R, b3R, Lm, Rm);
  init_kernel<<<dim3(2), dim3(256), 0, stream>>>(Lm, Rm, A0);
  lc_kernel<<<dim3(64), dim3(256), 0, stream>>>(Lm, Rm, colors, LC, RC);

  float* Ain  = A0;
  float* Aout = A1;
  int M = 2, wshift = 1;  // A shape (1,2) after depth 0

  // depths 1..3: tiny guarded path (M = 2, 4, 8)
  for (int d = 1; d <= 3; ++d) {
    expand_small<<<dim3(1, 4), dim3(256), 0, stream>>>(Ain, Aout, Lm, Rm, M,
                                                       wshift, d & 1);
    if (!(d & 1)) wshift += 1;
    M <<= 1;
    float* t = Ain; Ain = Aout; Aout = t;
  }

  // depths 4..14: fast path (M multiple of 16, W >= 8)
  for (int d = 4; d <= 14; ++d) {
    int strips = M >> 4;
    if (d & 1)
      expand_fast<1><<<dim3(strips, 4), dim3(256), 0, stream>>>(Ain, Aout, Lm,
                                                                Rm, wshift);
    else
      expand_fast<0><<<dim3(strips, 4), dim3(256), 0, stream>>>(Ain, Aout, Lm,
                                                                Rm, wshift);
    if (!(d & 1)) wshift += 1;
    M <<= 1;
    float* t = Ain; Ain = Aout; Aout = t;
  }

  // depth 15 (odd, wshift = 8) fused with color GEMM: M = 32768
  final_kernel<<<dim3(M >> 4), dim3(128), 0, stream>>>(Ain, LC, RC, out,
                                                       wshift);
}